// Transformer_2525440770897
// MI455X (gfx1250) — compile-verified
//
#include <hip/hip_runtime.h>
#include <hip/hip_bf16.h>
#include <math.h>

#define EMBED 128
#define HEADS 8
#define NX 3
#define FF 512
#define BATCH 512
#define CITY 100
#define ROWS (BATCH * CITY)          // 51200
#define S_ACT (ROWS * EMBED)         // 6,553,600 floats per activation buffer

typedef __attribute__((ext_vector_type(2))) float v2f;
typedef __attribute__((ext_vector_type(8))) float v8f;

static __device__ __forceinline__ v8f wmma4(v2f a, v2f b, v8f c) {
  // V_WMMA_F32_16X16X4_F32 : D(16x16 f32) = A(16x4) * B(4x16) + C
  return __builtin_amdgcn_wmma_f32_16x16x4_f32(false, a, false, b, (short)0, c, false, false);
}

// ---------------------------------------------------------------------------
// K1: embedding  xe[row, j] = x[row,0]*W[0,j] + x[row,1]*W[1,j] + b[j]
// ---------------------------------------------------------------------------
__global__ void embed_k(const float* __restrict__ x, const float* __restrict__ W,
                        const float* __restrict__ b, float* __restrict__ xe) {
  int idx = blockIdx.x * blockDim.x + threadIdx.x;
  if (idx >= ROWS * EMBED) return;
  int row = idx >> 7;
  int j = idx & 127;
  xe[idx] = x[row * 2] * W[j] + x[row * 2 + 1] * W[EMBED + j] + b[j];
}

// ---------------------------------------------------------------------------
// K2: WMMA fp32 GEMM  C[M,N] = A[M,K] @ B[K,N] (+bias)(+relu)
// one wave per 32x64 C tile: 2 A fragments share 4 B fragments -> 8 WMMA/k-step
// grid (M/32, N/64), block 32
// ---------------------------------------------------------------------------
__global__ void gemm32x64_k(const float* __restrict__ A, const float* __restrict__ B,
                            const float* __restrict__ bias, float* __restrict__ C,
                            int M, int N, int K, int relu) {
  int lane = threadIdx.x;                 // 0..31
  int m0 = blockIdx.x * 32;
  int n0 = blockIdx.y * 64;
  int aRow0 = m0 + (lane & 15);
  int kOff = (lane >> 4) * 2;             // lanes0-15 -> K=k,k+1 ; lanes16-31 -> K=k+2,k+3
  int bCol = n0 + (lane & 15);

  v8f acc00 = {}, acc01 = {}, acc02 = {}, acc03 = {};
  v8f acc10 = {}, acc11 = {}, acc12 = {}, acc13 = {};
  for (int k = 0; k < K; k += 4) {
    v2f a0, a1;
    const float* Ar0 = A + (long)aRow0 * K + k + kOff;
    const float* Ar1 = Ar0 + (long)16 * K;
    a0.x = Ar0[0]; a0.y = Ar0[1];
    a1.x = Ar1[0]; a1.y = Ar1[1];
    const float* B0 = B + (long)(k + kOff) * N;
    const float* B1 = B + (long)(k + kOff + 1) * N;
    v2f b0, b1, b2, b3;
    b0.x = B0[bCol];      b0.y = B1[bCol];
    b1.x = B0[bCol + 16]; b1.y = B1[bCol + 16];
    b2.x = B0[bCol + 32]; b2.y = B1[bCol + 32];
    b3.x = B0[bCol + 48]; b3.y = B1[bCol + 48];
    acc00 = wmma4(a0, b0, acc00);
    acc01 = wmma4(a0, b1, acc01);
    acc02 = wmma4(a0, b2, acc02);
    acc03 = wmma4(a0, b3, acc03);
    acc10 = wmma4(a1, b0, acc10);
    acc11 = wmma4(a1, b1, acc11);
    acc12 = wmma4(a1, b2, acc12);
    acc13 = wmma4(a1, b3, acc13);
  }
  int cCol = lane & 15;
  int rOff = (lane >> 4) * 8;             // lanes0-15 rows 0..7, lanes16-31 rows 8..15
  v8f accs[2][4] = {{acc00, acc01, acc02, acc03}, {acc10, acc11, acc12, acc13}};
#pragma unroll
  for (int t = 0; t < 2; ++t) {
#pragma unroll
    for (int j = 0; j < 4; ++j) {
      int col = n0 + 16 * j + cCol;
      float bi = bias ? bias[col] : 0.f;
#pragma unroll
      for (int r = 0; r < 8; ++r) {
        float v = accs[t][j][r] + bi;
        if (relu) v = fmaxf(v, 0.f);
        C[(long)(m0 + 16 * t + rOff + r) * N + col] = v;
      }
    }
  }
}

// ---------------------------------------------------------------------------
// K3: encoder self-attention, one block (128 thr) per (batch, head)
// Q,K,V: [row,128] with head h at cols 16h..16h+15.  O likewise.
// ---------------------------------------------------------------------------
__global__ void enc_attn_k(const float* __restrict__ Q, const float* __restrict__ K,
                           const float* __restrict__ V, float* __restrict__ O) {
  __shared__ float Qs[CITY * 16], Ks[CITY * 16], Vs[CITY * 16];
  __shared__ float sc[128], tmp[128];
  int b = blockIdx.x, h = blockIdx.y, tid = threadIdx.x;
  for (int i = tid; i < CITY * 16; i += 128) {
    int c = i >> 4, d = i & 15;
    long g = (long)(b * CITY + c) * EMBED + h * 16 + d;
    Qs[i] = Q[g]; Ks[i] = K[g]; Vs[i] = V[g];
  }
  __syncthreads();
  for (int q = 0; q < CITY; ++q) {
    float s = -1e30f;
    if (tid < CITY) {
      float d = 0.f;
#pragma unroll
      for (int i = 0; i < 16; ++i) d += Qs[q * 16 + i] * Ks[tid * 16 + i];
      s = d * 0.25f;                       // 1/sqrt(16)
    }
    sc[tid] = s; tmp[tid] = s;
    __syncthreads();
    for (int off = 64; off > 0; off >>= 1) {
      if (tid < off) tmp[tid] = fmaxf(tmp[tid], tmp[tid + off]);
      __syncthreads();
    }
    float mx = tmp[0];
    __syncthreads();
    float e = (tid < CITY) ? __expf(sc[tid] - mx) : 0.f;
    sc[tid] = e; tmp[tid] = e;
    __syncthreads();
    for (int off = 64; off > 0; off >>= 1) {
      if (tid < off) tmp[tid] += tmp[tid + off];
      __syncthreads();
    }
    float Z = tmp[0];
    __syncthreads();
    int g = tid >> 4, d = tid & 15;
    float acc = 0.f;
    for (int k = g; k < CITY; k += 8) acc += sc[k] * Vs[k * 16 + d];
    tmp[tid] = acc;
    __syncthreads();
    if (tid < 16) {
      float o = 0.f;
#pragma unroll
      for (int gg = 0; gg < 8; ++gg) o += tmp[gg * 16 + tid];
      O[(long)(b * CITY + q) * EMBED + h * 16 + tid] = o / Z;
    }
    __syncthreads();
  }
}

// ---------------------------------------------------------------------------
// K4: out = LN(X + Y) * g + b    (one wave per 128-wide row)
// ---------------------------------------------------------------------------
__global__ void add_ln_k(const float* __restrict__ X, const float* __restrict__ Y,
                         const float* __restrict__ g, const float* __restrict__ bta,
                         float* __restrict__ Out) {
  int wave = threadIdx.x >> 5, lane = threadIdx.x & 31;
  long row = (long)blockIdx.x * 8 + wave;
  float v[4], s = 0.f, s2 = 0.f;
#pragma unroll
  for (int i = 0; i < 4; ++i) {
    int c = lane + 32 * i;
    float t = X[row * EMBED + c] + Y[row * EMBED + c];
    v[i] = t; s += t; s2 += t * t;
  }
#pragma unroll
  for (int off = 16; off > 0; off >>= 1) {
    s += __shfl_xor(s, off, 32);
    s2 += __shfl_xor(s2, off, 32);
  }
  float m = s * (1.f / 128.f);
  float var = s2 * (1.f / 128.f) - m * m;
  float inv = rsqrtf(var + 1e-5f);
#pragma unroll
  for (int i = 0; i < 4; ++i) {
    int c = lane + 32 * i;
    Out[row * EMBED + c] = g[c] * (v[i] - m) * inv + bta[c];
  }
}

// ---------------------------------------------------------------------------
// K5: fused FF:  F = relu(H@W1 + b1) @ W2 + b2   (one block = 16 rows, 4 waves)
// ---------------------------------------------------------------------------
#define HS_STR 132
#define TS_STR 516
__global__ void ff_fused_k(const float* __restrict__ H, const float* __restrict__ W1,
                           const float* __restrict__ b1, const float* __restrict__ W2,
                           const float* __restrict__ b2, float* __restrict__ F) {
  __shared__ float Hs[16 * HS_STR];
  __shared__ float Ts[16 * TS_STR];
  int tid = threadIdx.x;                 // 128
  long m0 = (long)blockIdx.x * 16;
  for (int i = tid; i < 16 * EMBED; i += 128) {
    int r = i >> 7, c = i & 127;
    Hs[r * HS_STR + c] = H[(m0 + r) * EMBED + c];
  }
  __syncthreads();
  int lane = tid & 31, wave = tid >> 5;
  int aM = lane & 15, kOff = (lane >> 4) * 2, cCol = lane & 15, rOff = (lane >> 4) * 8;
  // stage 1: T[16,512] = relu(H@W1 + b1); each wave does 2 chunks of 64 cols
  for (int ch = wave; ch < 8; ch += 4) {
    int n0 = ch * 64;
    v8f acc[4] = {{}, {}, {}, {}};
    for (int k = 0; k < EMBED; k += 4) {
      v2f a;
      a.x = Hs[aM * HS_STR + k + kOff];
      a.y = Hs[aM * HS_STR + k + kOff + 1];
      const float* B0 = W1 + (k + kOff) * FF + n0;
      const float* B1 = W1 + (k + kOff + 1) * FF + n0;
#pragma unroll
      for (int j = 0; j < 4; ++j) {
        v2f bb; bb.x = B0[16 * j + cCol]; bb.y = B1[16 * j + cCol];
        acc[j] = wmma4(a, bb, acc[j]);
      }
    }
#pragma unroll
    for (int j = 0; j < 4; ++j) {
      int col = n0 + 16 * j + cCol;
      float bi = b1[col];
#pragma unroll
      for (int r = 0; r < 8; ++r)
        Ts[(rOff + r) * TS_STR + col] = fmaxf(acc[j][r] + bi, 0.f);
    }
  }
  __syncthreads();
  // stage 2: F[16,128] = T@W2 + b2; wave w -> cols [32w, 32w+32)
  int n0 = wave * 32;
  v8f acc0 = {}, acc1 = {};
  for (int k = 0; k < FF; k += 4) {
    v2f a;
    a.x = Ts[aM * TS_STR + k + kOff];
    a.y = Ts[aM * TS_STR + k + kOff + 1];
    const float* B0 = W2 + (k + kOff) * EMBED + n0;
    const float* B1 = W2 + (k + kOff + 1) * EMBED + n0;
    v2f bb;
    bb.x = B0[cCol];      bb.y = B1[cCol];      acc0 = wmma4(a, bb, acc0);
    bb.x = B0[16 + cCol]; bb.y = B1[16 + cCol]; acc1 = wmma4(a, bb, acc1);
  }
  v8f accs[2] = {acc0, acc1};
#pragma unroll
  for (int j = 0; j < 2; ++j) {
    int col = n0 + 16 * j + cCol;
    float bi = b2[col];
#pragma unroll
    for (int r = 0; r < 8; ++r)
      F[(m0 + rOff + r) * EMBED + col] = accs[j][r] + bi;
  }
}

// ---------------------------------------------------------------------------
// K6: enc_mean[b,j] = mean_c xe[b,c,j]
// ---------------------------------------------------------------------------
__global__ void mean_k(const float* __restrict__ xe, float* __restrict__ mean) {
  int b = blockIdx.x, j = threadIdx.x;
  float s = 0.f;
  for (int c = 0; c < CITY; ++c) s += xe[(long)(b * CITY + c) * EMBED + j];
  mean[b * EMBED + j] = s * (1.f / (float)CITY);
}

// ---------------------------------------------------------------------------
// K7: decoder state init
// ---------------------------------------------------------------------------
__global__ void dec_init_k(const float* __restrict__ v1, const float* __restrict__ vf,
                           float* __restrict__ h0, float* __restrict__ hprev,
                           float* __restrict__ mask, float* __restrict__ logl) {
  int b = blockIdx.x, tid = threadIdx.x;
  h0[b * EMBED + tid] = v1[tid];
  hprev[b * EMBED + tid] = vf[tid];
  if (tid < CITY) mask[b * CITY + tid] = 0.f;
  if (tid == 0) logl[b] = 0.f;
}

// ---------------------------------------------------------------------------
// K8: one greedy decode step (one block of 128 thr per batch element)
// ---------------------------------------------------------------------------
__global__ void dec_step_k(int step, const float* __restrict__ mean,
                           float* __restrict__ h0, float* __restrict__ hprev,
                           const float* __restrict__ enc_out, const float* __restrict__ Kd,
                           const float* __restrict__ Vd, const float* __restrict__ kl,
                           float* __restrict__ mask, float* __restrict__ logl,
                           int* __restrict__ tour,
                           const float* __restrict__ Wctx, const float* __restrict__ Wq,
                           const float* __restrict__ Wo) {
  __shared__ float ms[128], h0s[128], hps[128], ctxs[128], qs[128], atts[128], gs[128];
  __shared__ float sc[128], tmp[128];
  __shared__ int tmpi[128];
  __shared__ int nxt_s;
  int b = blockIdx.x, tid = threadIdx.x;

  ms[tid] = mean[b * EMBED + tid];
  h0s[tid] = h0[b * EMBED + tid];
  hps[tid] = hprev[b * EMBED + tid];
  float mk = (tid < CITY) ? mask[b * CITY + tid] : 0.f;
  __syncthreads();

  // ctx = [mean, h0, hprev] @ Wctx
  float c = 0.f;
  for (int k = 0; k < 128; ++k)
    c += ms[k] * Wctx[k * 128 + tid] + h0s[k] * Wctx[(128 + k) * 128 + tid]
       + hps[k] * Wctx[(256 + k) * 128 + tid];
  ctxs[tid] = c;
  __syncthreads();
  // q = ctx @ Wq
  float qv = 0.f;
  for (int k = 0; k < 128; ++k) qv += ctxs[k] * Wq[k * 128 + tid];
  qs[tid] = qv;
  __syncthreads();

  // masked glimpse attention, per head
  for (int h = 0; h < HEADS; ++h) {
    float s = -1e30f;
    if (tid < CITY) {
      float d = 0.f;
#pragma unroll
      for (int i = 0; i < 16; ++i) d += qs[h * 16 + i] * Kd[(long)(b * CITY + tid) * EMBED + h * 16 + i];
      s = d * 0.25f + mk * (-1e9f);
    }
    sc[tid] = s; tmp[tid] = s;
    __syncthreads();
    for (int off = 64; off > 0; off >>= 1) {
      if (tid < off) tmp[tid] = fmaxf(tmp[tid], tmp[tid + off]);
      __syncthreads();
    }
    float mx = tmp[0];
    __syncthreads();
    float e = (tid < CITY) ? __expf(sc[tid] - mx) : 0.f;
    sc[tid] = e; tmp[tid] = e;
    __syncthreads();
    for (int off = 64; off > 0; off >>= 1) {
      if (tid < off) tmp[tid] += tmp[tid + off];
      __syncthreads();
    }
    float Z = tmp[0];
    __syncthreads();
    int g = tid >> 4, d = tid & 15;
    float acc = 0.f;
    for (int k = g; k < CITY; k += 8) acc += sc[k] * Vd[(long)(b * CITY + k) * EMBED + h * 16 + d];
    tmp[tid] = acc;
    __syncthreads();
    if (tid < 16) {
      float o = 0.f;
#pragma unroll
      for (int gg = 0; gg < 8; ++gg) o += tmp[gg * 16 + tid];
      atts[h * 16 + tid] = o / Z;
    }
    __syncthreads();
  }

  // glimpse = att @ Wo
  float gv = 0.f;
  for (int k = 0; k < 128; ++k) gv += atts[k] * Wo[k * 128 + tid];
  gs[tid] = gv;
  __syncthreads();

  // pointer logits
  float lg = -1e30f;
  if (tid < CITY) {
    float d = 0.f;
    for (int k = 0; k < 128; ++k) d += gs[k] * kl[(long)(b * CITY + tid) * EMBED + k];
    lg = 10.f * tanhf(d * 0.08838834764831845f) + mk * (-1e9f);  // /sqrt(128)
  }
  sc[tid] = lg; tmp[tid] = lg;
  tmpi[tid] = (tid < CITY) ? tid : (1 << 30);
  __syncthreads();
  for (int off = 64; off > 0; off >>= 1) {
    if (tid < off) {
      float v2 = tmp[tid + off]; int i2 = tmpi[tid + off];
      if (v2 > tmp[tid] || (v2 == tmp[tid] && i2 < tmpi[tid])) { tmp[tid] = v2; tmpi[tid] = i2; }
    }
    __syncthreads();
  }
  float mx = tmp[0];
  int nxt = tmpi[0];
  __syncthreads();
  float e = (tid < CITY) ? __expf(sc[tid] - mx) : 0.f;
  tmp[tid] = e;
  __syncthreads();
  for (int off = 64; off > 0; off >>= 1) {
    if (tid < off) tmp[tid] += tmp[tid + off];
    __syncthreads();
  }
  float Z = tmp[0];
  if (tid == 0) {
    nxt_s = nxt;
    logl[b] += -__logf(Z);                 // logits[nxt] == mx -> lp = -log(sum exp)
    tour[b * CITY + step] = nxt;
    mask[b * CITY + nxt] += 1.f;
  }
  __syncthreads();
  int nx = nxt_s;
  float gval = enc_out[(long)(b * CITY + nx) * EMBED + tid];
  hprev[b * EMBED + tid] = gval;
  if (step == 0) h0[b * EMBED + tid] = gval;
}

// ---------------------------------------------------------------------------
// K9: final gather -> d_out = [out_tour (512*100*2 f32), log_l (512 f32)]
// ---------------------------------------------------------------------------
__global__ void gather_k(const float* __restrict__ x, const int* __restrict__ tour,
                         const float* __restrict__ logl, float* __restrict__ out) {
  int idx = blockIdx.x * blockDim.x + threadIdx.x;
  if (idx < ROWS) {
    int b = idx / CITY;
    int n = tour[idx];
    out[idx * 2]     = x[(long)(b * CITY + n) * 2];
    out[idx * 2 + 1] = x[(long)(b * CITY + n) * 2 + 1];
  }
  if (idx < BATCH) out[ROWS * 2 + idx] = logl[idx];
}

// ---------------------------------------------------------------------------
extern "C" void kernel_launch(void* const* d_in, const int* in_sizes, int n_in,
                              void* d_out, int out_size, void* d_ws, size_t ws_size,
                              hipStream_t stream) {
  (void)in_sizes; (void)n_in; (void)out_size; (void)ws_size;
  const float* x      = (const float*)d_in[0];
  const float* W_emb  = (const float*)d_in[1];
  const float* b_emb  = (const float*)d_in[2];
  const float* eWq    = (const float*)d_in[3];
  const float* eWk    = (const float*)d_in[4];
  const float* eWv    = (const float*)d_in[5];
  const float* eWo    = (const float*)d_in[6];
  const float* ln1g   = (const float*)d_in[7];
  const float* ln1b   = (const float*)d_in[8];
  const float* ln2g   = (const float*)d_in[9];
  const float* ln2b   = (const float*)d_in[10];
  const float* W1     = (const float*)d_in[11];
  const float* b1     = (const float*)d_in[12];
  const float* W2     = (const float*)d_in[13];
  const float* b2     = (const float*)d_in[14];
  const float* v1     = (const float*)d_in[15];
  const float* vf     = (const float*)d_in[16];
  const float* dWctx  = (const float*)d_in[17];
  const float* dWq    = (const float*)d_in[18];
  const float* dWk    = (const float*)d_in[19];
  const float* dWv    = (const float*)d_in[20];
  const float* dWo    = (const float*)d_in[21];
  const float* dWkl   = (const float*)d_in[22];

  float* w = (float*)d_ws;
  float* bufA = w;               // h / enc_out
  float* bufB = w + (long)S_ACT; // attention out / FF out
  float* bufQ = w + 2L * S_ACT;  // Q, later kl
  float* bufK = w + 3L * S_ACT;  // K, later Kd
  float* bufV = w + 4L * S_ACT;  // V, later Vd
  float* meanb = w + 5L * S_ACT;
  float* h0    = meanb + BATCH * EMBED;
  float* hprev = h0 + BATCH * EMBED;
  float* maskb = hprev + BATCH * EMBED;
  float* loglb = maskb + BATCH * CITY;
  int*   tourb = (int*)(loglb + BATCH);

  dim3 g2(ROWS / 32, EMBED / 64);     // (1600, 2)

  // embedding + enc_mean (reference uses x_embed mean)
  embed_k<<<(ROWS * EMBED + 255) / 256, 256, 0, stream>>>(x, W_emb, b_emb, bufA);
  mean_k<<<BATCH, 128, 0, stream>>>(bufA, meanb);

  // encoder layers
  for (int i = 0; i < NX; ++i) {
    const float* Wq = eWq + (long)i * EMBED * EMBED;
    const float* Wk = eWk + (long)i * EMBED * EMBED;
    const float* Wv = eWv + (long)i * EMBED * EMBED;
    const float* Wo = eWo + (long)i * EMBED * EMBED;
    gemm32x64_k<<<g2, 32, 0, stream>>>(bufA, Wq, nullptr, bufQ, ROWS, EMBED, EMBED, 0);
    gemm32x64_k<<<g2, 32, 0, stream>>>(bufA, Wk, nullptr, bufK, ROWS, EMBED, EMBED, 0);
    gemm32x64_k<<<g2, 32, 0, stream>>>(bufA, Wv, nullptr, bufV, ROWS, EMBED, EMBED, 0);
    enc_attn_k<<<dim3(BATCH, HEADS), 128, 0, stream>>>(bufQ, bufK, bufV, bufB);
    gemm32x64_k<<<g2, 32, 0, stream>>>(bufB, Wo, nullptr, bufQ, ROWS, EMBED, EMBED, 0);
    add_ln_k<<<ROWS / 8, 256, 0, stream>>>(bufA, bufQ, ln1g + i * EMBED, ln1b + i * EMBED, bufA);
    ff_fused_k<<<ROWS / 16, 128, 0, stream>>>(bufA, W1 + (long)i * EMBED * FF, b1 + i * FF,
                                              W2 + (long)i * FF * EMBED, b2 + i * EMBED, bufB);
    add_ln_k<<<ROWS / 8, 256, 0, stream>>>(bufA, bufB, ln2g + i * EMBED, ln2b + i * EMBED, bufA);
  }

  // decoder precompute: Kd, Vd, kl (enc_out is constant across steps)
  gemm32x64_k<<<g2, 32, 0, stream>>>(bufA, dWk, nullptr, bufK, ROWS, EMBED, EMBED, 0);
  gemm32x64_k<<<g2, 32, 0, stream>>>(bufA, dWv, nullptr, bufV, ROWS, EMBED, EMBED, 0);
  gemm32x64_k<<<g2, 32, 0, stream>>>(bufA, dWkl, nullptr, bufQ, ROWS, EMBED, EMBED, 0);

  dec_init_k<<<BATCH, 128, 0, stream>>>(v1, vf, h0, hprev, maskb, loglb);
  for (int step = 0; step < CITY; ++step) {
    dec_step_k<<<BATCH, 128, 0, stream>>>(step, meanb, h0, hprev, bufA, bufK, bufV, bufQ,
                                          maskb, loglb, tourb, dWctx, dWq, dWo);
  }
  gather_k<<<(ROWS + 255) / 256, 256, 0, stream>>>(x, tourb, loglb, (float*)d_out);
}